// ImprovedDetectionLoss_40948218200551
// MI455X (gfx1250) — compile-verified
//
#include <hip/hip_runtime.h>
#include <cstdint>
#include <cstddef>

typedef unsigned long long u64;
typedef unsigned int u32;
typedef __attribute__((ext_vector_type(2))) float v2f;
typedef __attribute__((ext_vector_type(8))) float v8f;

#define BB 32
#define NN 65536
#define MM 64
#define CHUNK 2048
#define CPI (NN / CHUNK)      // 32 chunks per image
#define KPT (CHUNK / 256)     // 8 preds per thread
#define POS_THRF 0.5f
#define NEG_THRF 0.4f
#define SCAT_THRF 0.1f
#define ALPHAF 0.25f
#define BBOX_WF 2.0f
#define EPSF 1e-6f

// workspace layout (bytes)
#define WIN_OFF 0u
#define WIN_BYTES (BB * MM * 8u)          // 16384
#define POS_OFF (WIN_OFF + WIN_BYTES)     // 16384
#define POS_WORDS (BB * NN / 32)          // 65536 u32
#define POS_BYTES (POS_WORDS * 4u)        // 262144
#define ACC_OFF (POS_OFF + POS_BYTES)     // 278528
#define ACC_BYTES (BB * 4u * 4u)          // 512
#define PP_OFF (ACC_OFF + ACC_BYTES)      // 279040 (8B aligned)
#define PP_BYTES ((size_t)BB * (size_t)NN * 8ull) // 16 MiB per-pred packed best
#define BASE_BYTES PP_OFF

// ---------- CDNA5 async global->LDS staging (ASYNCcnt path) ----------
// LDS dest offset from a VGPR; global addr = 64-bit SGPR base + 32-bit VGPR offset
__device__ __forceinline__ void async_load_b128_to_lds(u32 lds_off,
                                                       const void* sbase,
                                                       u32 voff) {
  asm volatile("global_load_async_to_lds_b128 %0, %1, %2"
               :
               : "v"(lds_off), "v"(voff), "s"(sbase)
               : "memory");
}
__device__ __forceinline__ void wait_asynccnt0() {
#if defined(__has_builtin)
#if __has_builtin(__builtin_amdgcn_s_wait_asynccnt)
  __builtin_amdgcn_s_wait_asynccnt(0);
  return;
#endif
#endif
  asm volatile("s_wait_asynccnt 0" ::: "memory");
}
// generic pointer to a __shared__ object: low 32 bits are the LDS offset
__device__ __forceinline__ u32 lds_off_of(const void* p) {
  return (u32)(uintptr_t)p;
}

// ---------- math helpers ----------
struct Box { float x1, y1, x2, y2; };

__device__ __forceinline__ float iou_fn(Box p, Box g) {
  float ix1 = fmaxf(p.x1, g.x1), iy1 = fmaxf(p.y1, g.y1);
  float ix2 = fminf(p.x2, g.x2), iy2 = fminf(p.y2, g.y2);
  float inter = fmaxf(ix2 - ix1, 0.0f) * fmaxf(iy2 - iy1, 0.0f);
  float ap = (p.x2 - p.x1) * (p.y2 - p.y1);
  float ag = (g.x2 - g.x1) * (g.y2 - g.y1);
  float un = ap + ag - inter;
  return inter / fmaxf(un, EPSF);
}

__device__ __forceinline__ float giou_term(Box p, Box g) {
  float ix1 = fmaxf(p.x1, g.x1), iy1 = fmaxf(p.y1, g.y1);
  float ix2 = fminf(p.x2, g.x2), iy2 = fminf(p.y2, g.y2);
  float inter = fmaxf(ix2 - ix1, 0.0f) * fmaxf(iy2 - iy1, 0.0f);
  float ap = (p.x2 - p.x1) * (p.y2 - p.y1);
  float ag = (g.x2 - g.x1) * (g.y2 - g.y1);
  float un = ap + ag - inter;
  float iou = inter / fmaxf(un, EPSF);
  float ex1 = fminf(p.x1, g.x1), ey1 = fminf(p.y1, g.y1);
  float ex2 = fmaxf(p.x2, g.x2), ey2 = fmaxf(p.y2, g.y2);
  float enc = (ex2 - ex1) * (ey2 - ey1);
  float giou = iou - (enc - un) / fmaxf(enc, EPSF);
  return 1.0f - giou;
}

// ---------- kernels ----------
__global__ void k_init(u32* __restrict__ base_ws) {
  u32 i = blockIdx.x * blockDim.x + threadIdx.x;
  if (i < (BASE_BYTES / 4u)) base_ws[i] = 0u;
}

__global__ void __launch_bounds__(256)
k_phase1(const float* __restrict__ pred, const float* __restrict__ gt,
         u64* __restrict__ winners, u64* __restrict__ perpred) {
  __shared__ float4 sp[CHUNK];          // staged pred boxes (32 KB)
  __shared__ float4 sgt[MM];            // staged GT boxes (1 KB, async DMA'd)
  __shared__ u64 swin[MM];              // per-GT block winners

  const int t = threadIdx.x;
  const int b = blockIdx.x / CPI;
  const int chunk = blockIdx.x % CPI;
  const int p0 = chunk * CHUNK;

  // stage GT boxes for this image via CDNA5 async global->LDS (128b/lane)
  if (t < MM) {
    const float* gbase = gt + (size_t)b * MM * 4;
    async_load_b128_to_lds(lds_off_of(&sgt[t]), gbase, (u32)t * 16u);
  }

  // stage pred boxes (also kept in registers for Part A)
  Box pb[KPT];
  for (int i = 0; i < KPT; ++i) {
    int lp = t + 256 * i;
    size_t base = ((size_t)b * NN + (size_t)(p0 + lp)) * 5;
    float cx = pred[base + 0], cy = pred[base + 1];
    float w  = pred[base + 2], h  = pred[base + 3];
    Box bx{cx - 0.5f * w, cy - 0.5f * h, cx + 0.5f * w, cy + 0.5f * h};
    pb[i] = bx;
    sp[lp] = make_float4(bx.x1, bx.y1, bx.x2, bx.y2);
  }
  wait_asynccnt0();
  __syncthreads();

  // Part A: per-pred best GT (first-argmax via strict >, ascending j)
  if (perpred != nullptr) {
    for (int i = 0; i < KPT; ++i) {
      float bestv = -1.0f; int bestj = 0;
      for (int j = 0; j < MM; ++j) {
        float4 g4 = sgt[j];
        Box g{g4.x, g4.y, g4.z, g4.w};
        float v = iou_fn(pb[i], g);
        if (v > bestv) { bestv = v; bestj = j; }
      }
      int lp = t + 256 * i;
      perpred[(size_t)b * NN + (size_t)(p0 + lp)] =
          ((u64)__float_as_uint(bestv) << 32) | (u64)(u32)bestj;
    }
  }

  // Part B: per-GT argmax. Lane layout: jj = t&63 (GT), sl = t>>6 (pred slice).
  // All lanes of a wave share the same slice -> broadcast LDS reads of sp[].
  const int jj = t & (MM - 1);
  const int sl = t >> 6;
  float4 g4 = sgt[jj];
  Box g{g4.x, g4.y, g4.z, g4.w};
  u64 best = 0ull;
  for (int i = 0; i < CHUNK / 4; ++i) {
    int lp = 4 * i + sl;
    float4 p4 = sp[lp];
    Box p{p4.x, p4.y, p4.z, p4.w};
    float v = iou_fn(p, g);
    // pack (iou_bits << 32) | (~pred_idx): max -> highest IoU, first index on ties
    u64 cand = ((u64)__float_as_uint(v) << 32) |
               (u64)(0xFFFFFFFFu - (u32)(p0 + lp));
    best = (cand > best) ? cand : best;
  }
  if (sl == 0) swin[jj] = best;
  __syncthreads();
  if (sl != 0) atomicMax(&swin[jj], best);
  __syncthreads();
  if (t < MM) atomicMax(&winners[b * MM + t], swin[t]);
}

__global__ void k_scatter(const u64* __restrict__ winners,
                          u32* __restrict__ posbits) {
  int t = blockIdx.x * blockDim.x + threadIdx.x;
  if (t >= BB * MM) return;
  int b = t >> 6;
  u64 v = winners[t];
  float val = __uint_as_float((u32)(v >> 32));
  u32 pidx = 0xFFFFFFFFu - (u32)(v & 0xFFFFFFFFull);
  if (val > SCAT_THRF && pidx < NN) {
    atomicOr(&posbits[((u32)b * NN + pidx) >> 5], 1u << (pidx & 31u));
  }
}

__global__ void __launch_bounds__(256)
k_phase2(const float* __restrict__ pred, const float* __restrict__ gt,
         const u64* __restrict__ perpred, const u32* __restrict__ posbits,
         float* __restrict__ acc) {
  __shared__ float4 sgt[MM];
  __shared__ float red[8][4];

  const int t = threadIdx.x;
  const int b = blockIdx.x / CPI;
  const int chunk = blockIdx.x % CPI;
  const int p0 = chunk * CHUNK;

  if (t < MM) {
    const float* gbase = gt + (size_t)b * MM * 4;
    async_load_b128_to_lds(lds_off_of(&sgt[t]), gbase, (u32)t * 16u);
  }
  wait_asynccnt0();
  __syncthreads();

  float cls = 0.0f, vc = 0.0f, pc = 0.0f, rg = 0.0f;
  for (int i = 0; i < KPT; ++i) {
    int lp = t + 256 * i;
    int p = p0 + lp;
    size_t base = ((size_t)b * NN + (size_t)p) * 5;
    if (i + 1 < KPT) __builtin_prefetch(&pred[base + 5 * 256], 0, 0);
    float cx = pred[base + 0], cy = pred[base + 1];
    float w  = pred[base + 2], h  = pred[base + 3];
    float l  = pred[base + 4];
    Box pbx{cx - 0.5f * w, cy - 0.5f * h, cx + 0.5f * w, cy + 0.5f * h};

    float bestv; int bestj;
    if (perpred != nullptr) {
      u64 v = perpred[(size_t)b * NN + (size_t)p];
      bestv = __uint_as_float((u32)(v >> 32));
      bestj = (int)(v & 63ull);
    } else {
      bestv = -1.0f; bestj = 0;
      for (int j = 0; j < MM; ++j) {
        float4 g4 = sgt[j];
        Box g{g4.x, g4.y, g4.z, g4.w};
        float v = iou_fn(pbx, g);
        if (v > bestv) { bestv = v; bestj = j; }
      }
    }

    u32 bits = posbits[((u32)b * NN + (u32)p) >> 5];
    bool scat = (bits >> (p & 31)) & 1u;
    bool pos = scat || (bestv > POS_THRF);
    bool neg = bestv < NEG_THRF;
    bool valid = pos || neg;

    float tt = pos ? 1.0f : 0.0f;
    float bce = fmaxf(l, 0.0f) - l * tt + log1pf(expf(-fabsf(l)));
    float pr = 1.0f / (1.0f + expf(-l));
    float pt = pos ? pr : 1.0f - pr;
    float at = pos ? ALPHAF : 1.0f - ALPHAF;
    float fl = at * (1.0f - pt) * (1.0f - pt) * bce;

    cls += valid ? fl : 0.0f;
    vc  += valid ? 1.0f : 0.0f;
    pc  += tt;
    float4 g4 = sgt[bestj];
    Box g{g4.x, g4.y, g4.z, g4.w};
    rg += pos ? giou_term(pbx, g) : 0.0f;
  }

  // intra-wave (wave32) shuffle reduction
  for (int o = 16; o > 0; o >>= 1) {
    cls += __shfl_down(cls, o, 32);
    vc  += __shfl_down(vc, o, 32);
    pc  += __shfl_down(pc, o, 32);
    rg  += __shfl_down(rg, o, 32);
  }
  int wid = t >> 5, lid = t & 31;
  if (lid == 0) { red[wid][0] = cls; red[wid][1] = vc; red[wid][2] = rg; red[wid][3] = pc; }
  __syncthreads();

  // Cross-wave reduction via V_WMMA_F32_16X16X4_F32 (fp32 tensor path):
  // D = ones(16x4) x B(4x16):  D[m][n] = sum_k B[k][n].
  // B[k][n] holds red[k + 4*(n>=4)][n&3]  (n<8), zero elsewhere, so
  // column c sums waves 0..3 and column 4+c sums waves 4..7 for channel c.
  // B VGPR layout (4x16, 32-bit): lanes 0-15: {B[0][l], B[1][l]},
  //                               lanes 16-31: {B[2][l], B[3][l]}.
  if (t < 32) {                 // wave 0, EXEC all ones (uniform branch)
    int l = t & 15;
    int half = t >> 4;          // 0 -> K=0/1, 1 -> K=2/3
    float b0 = 0.0f, b1 = 0.0f;
    if (l < 8) {
      int w = ((l >> 2) << 2) + (half << 1);  // 0,2,4,6 base wave
      int c = l & 3;
      b0 = red[w][c];
      b1 = red[w + 1][c];
    }
    v2f Av; Av[0] = 1.0f; Av[1] = 1.0f;       // A = ones(16x4)
    v2f Bv; Bv[0] = b0;   Bv[1] = b1;
    v8f Cv = {};
    v8f Dv = __builtin_amdgcn_wmma_f32_16x16x4_f32(
        false, Av, false, Bv, (short)0, Cv, false, false);
    float d0 = Dv[0];                         // row M=0 (lanes 0-15): D[0][lane]
    float partner = __shfl(d0, (t & 3) + 4, 32);  // D[0][c+4]
    if (t < 4) atomicAdd(&acc[b * 4 + t], d0 + partner);
  }
}

__global__ void k_final(const float* __restrict__ acc, float* __restrict__ out) {
  int b = threadIdx.x;  // 32 lanes == one wave32 == B images
  float cn = acc[b * 4 + 0], vcnt = acc[b * 4 + 1];
  float rn = acc[b * 4 + 2], pcnt = acc[b * 4 + 3];
  float cls_b = (vcnt > 0.0f) ? cn / fmaxf(vcnt, 1.0f) : 0.0f;
  float reg_b = (pcnt > 0.0f) ? rn / fmaxf(pcnt, 1.0f) : 0.0f;
  float np = pcnt;
  for (int o = 16; o > 0; o >>= 1) {
    cls_b += __shfl_down(cls_b, o, 32);
    reg_b += __shfl_down(reg_b, o, 32);
    np    += __shfl_down(np, o, 32);
  }
  if (b == 0) {
    float num_pos = fmaxf(np, 1.0f);
    float total_cls = cls_b / (float)BB;
    float total_reg = reg_b / num_pos * (float)BB;
    out[0] = total_cls + BBOX_WF * total_reg;
  }
}

extern "C" void kernel_launch(void* const* d_in, const int* in_sizes, int n_in,
                              void* d_out, int out_size, void* d_ws, size_t ws_size,
                              hipStream_t stream) {
  const float* pred = (const float*)d_in[0];
  const float* gt   = (const float*)d_in[1];
  float* out = (float*)d_out;
  char* ws = (char*)d_ws;

  u64* winners = (u64*)(ws + WIN_OFF);
  u32* posbits = (u32*)(ws + POS_OFF);
  float* acc   = (float*)(ws + ACC_OFF);
  u64* perpred = (ws_size >= (size_t)PP_OFF + PP_BYTES) ? (u64*)(ws + PP_OFF)
                                                        : (u64*)nullptr;

  const u32 initWords = BASE_BYTES / 4u;
  k_init<<<(initWords + 255) / 256, 256, 0, stream>>>((u32*)ws);
  k_phase1<<<BB * CPI, 256, 0, stream>>>(pred, gt, winners, perpred);
  k_scatter<<<(BB * MM + 255) / 256, 256, 0, stream>>>(winners, posbits);
  k_phase2<<<BB * CPI, 256, 0, stream>>>(pred, gt, perpred, posbits, acc);
  k_final<<<1, 32, 0, stream>>>(acc, out);
}